// ClassMean_PT_29480655520055
// MI455X (gfx1250) — compile-verified
//
#include <hip/hip_runtime.h>
#include <hip/hip_bf16.h>
#include <stdint.h>

#define N_CLASSES 1000
#define D 128
#define SLICES 4
#define COLS 32            // columns per slice
#define ACC_STRIDE 33      // pad so LDS bank = (33*class + col) % 64 rotates per class
#define TPB 256
#define WAVES 8            // TPB / 32
#define J 16               // rows per thread per super-tile
#define ROWS_PER_TILE 512  // 32 row-groups * J
#define CHUNKS 64          // row-chunk blocks (grid.x) -> 256 blocks total

typedef float v4f __attribute__((ext_vector_type(4)));

// gfx1250 async global->LDS copies (ASYNCcnt-tracked).
__device__ __forceinline__ void async_copy_b64(uint32_t lds_off, uint64_t gaddr) {
    asm volatile("global_load_async_to_lds_b64 %0, %1, off"
                 :: "v"(lds_off), "v"(gaddr) : "memory");
}
__device__ __forceinline__ void wait_async_le1() {
    asm volatile("s_wait_asynccnt 0x1" ::: "memory");
}
__device__ __forceinline__ void wait_async_0() {
    asm volatile("s_wait_asynccnt 0x0" ::: "memory");
}

__global__ __launch_bounds__(TPB)
void class_mean_accum(const float* __restrict__ x,
                      const int* __restrict__ classes,
                      float* __restrict__ ws,
                      int N, int superTiles)
{
    __shared__ float sAcc[N_CLASSES * ACC_STRIDE];        // 132,000 B
    __shared__ float sCnt[N_CLASSES];                     //   4,000 B
    __shared__ alignas(16) int sCls[WAVES][2][2 * 32];    //   4,096 B (wave-private)

    const int t    = threadIdx.x;
    const int w    = t >> 5;          // wave id
    const int lane = t & 31;
    const int slice = blockIdx.y;
    const int sliceBase = slice * COLS;
    const int rid  = t >> 3;          // 0..31 : row-in-group
    const int q    = rid & 3;         // row-in-group within this wave (0..3)
    const int colq = (t & 7) * 4;     // float4 column within slice
    const bool cntLane = (slice == 0) && ((t & 7) == 0);

    for (int i = t; i < N_CLASSES * ACC_STRIDE; i += TPB) sAcc[i] = 0.0f;
    for (int i = t; i < N_CLASSES; i += TPB) sCnt[i] = 0.0f;
    __syncthreads();

    const int s0 = blockIdx.x;
    // Wave-private class staging: lane L of wave w copies the b64 pair
    // classes[base + srcOff .. +1]; exactly the rows this wave consumes.
    const int srcOff = 4 * w + 32 * (lane >> 1) + (lane & 1) * 2;

    // ---- prologue: stage classes (buf 0) + load x for tile s0 ----
    {
        int idx = s0 * ROWS_PER_TILE + srcOff;
        idx = idx < (N - 1) ? idx : (N - 2);
        async_copy_b64((uint32_t)(uintptr_t)&sCls[w][0][2 * lane],
                       (uint64_t)(uintptr_t)(classes + idx));
    }
    v4f vC[J];
    {
        const int base = s0 * ROWS_PER_TILE;
#pragma unroll
        for (int j = 0; j < J; ++j) {
            int row = base + rid + 32 * j;
            int rr = row < N ? row : (N - 1);
            vC[j] = __builtin_nontemporal_load(
                (const v4f*)(x + (size_t)rr * D + sliceBase + colq));
        }
    }

    int bufC = 0;
    for (int s = s0; s < superTiles; s += CHUNKS) {
        const int sn = s + CHUNKS;
        const int bufN = bufC ^ 1;

        // stage next tile's classes (clamped; harmless past the end)
        {
            int idx = sn * ROWS_PER_TILE + srcOff;
            idx = idx < (N - 1) ? idx : (N - 2);
            async_copy_b64((uint32_t)(uintptr_t)&sCls[w][bufN][2 * lane],
                           (uint64_t)(uintptr_t)(classes + idx));
        }
        // our wave's copies for bufC are done (wave-private => no barrier)
        wait_async_le1();

        int cls[J];
#pragma unroll
        for (int j = 0; j < J; ++j) cls[j] = sCls[w][bufC][4 * j + q];

        // issue next tile's x loads NOW so they overlap the atomic phase
        v4f vN[J];
        {
            const int nbase = sn * ROWS_PER_TILE;
#pragma unroll
            for (int j = 0; j < J; ++j) {
                int row = nbase + rid + 32 * j;
                int rr = row < N ? row : (N - 1);
                vN[j] = __builtin_nontemporal_load(
                    (const v4f*)(x + (size_t)rr * D + sliceBase + colq));
            }
        }

        // consume current tile: LDS-privatized segment accumulation
        const int base = s * ROWS_PER_TILE;
#pragma unroll
        for (int j = 0; j < J; ++j) {
            int row = base + rid + 32 * j;
            if (row < N) {
                float* a = &sAcc[cls[j] * ACC_STRIDE + colq];
                atomicAdd(a + 0, vC[j].x);
                atomicAdd(a + 1, vC[j].y);
                atomicAdd(a + 2, vC[j].z);
                atomicAdd(a + 3, vC[j].w);
                if (cntLane) atomicAdd(&sCnt[cls[j]], 1.0f);
            }
        }
#pragma unroll
        for (int j = 0; j < J; ++j) vC[j] = vN[j];
        bufC = bufN;
    }

    wait_async_0();   // drain speculative last prefetch
    __syncthreads();

    // flush LDS partials into L2-resident global accumulators (516 KB)
    float* wsSums = ws;
    float* wsCnt  = ws + N_CLASSES * D;
    for (int i = t; i < N_CLASSES * COLS; i += TPB) {
        int c = i >> 5, col = i & 31;
        atomicAdd(&wsSums[c * D + sliceBase + col], sAcc[c * ACC_STRIDE + col]);
    }
    if (slice == 0) {
        for (int c = t; c < N_CLASSES; c += TPB) {
            atomicAdd(&wsCnt[c], sCnt[c]);
        }
    }
}

__global__ __launch_bounds__(TPB)
void class_mean_finalize(const float* __restrict__ ws, float* __restrict__ out)
{
    int i = blockIdx.x * TPB + threadIdx.x;
    if (i < N_CLASSES * D) {
        int c = i >> 7;  // D == 128
        float cnt = ws[N_CLASSES * D + c];
        out[i] = ws[i] * (1.0f / cnt);
    }
}

extern "C" void kernel_launch(void* const* d_in, const int* in_sizes, int n_in,
                              void* d_out, int out_size, void* d_ws, size_t ws_size,
                              hipStream_t stream)
{
    const float* x       = (const float*)d_in[0];
    const int*   classes = (const int*)d_in[1];
    float*       out     = (float*)d_out;
    float*       ws      = (float*)d_ws;

    const int N = in_sizes[1];  // number of rows (2,000,000)
    const int superTiles = (N + ROWS_PER_TILE - 1) / ROWS_PER_TILE;

    hipMemsetAsync(d_ws, 0, (size_t)(N_CLASSES * D + N_CLASSES) * sizeof(float),
                   stream);

    dim3 grid(CHUNKS, SLICES);
    class_mean_accum<<<grid, TPB, 0, stream>>>(x, classes, ws, N, superTiles);

    int fin_blocks = (N_CLASSES * D + TPB - 1) / TPB;
    class_mean_finalize<<<fin_blocks, TPB, 0, stream>>>(ws, out);
}